// DGLFeatureGAT_23922967839172
// MI455X (gfx1250) — compile-verified
//
#include <hip/hip_runtime.h>

// GATv2 fused kernel for MI455X (gfx1250, wave32, WMMA + TDM).
// One block per (batch, head). All intermediates live in LDS.
// - x[b] tile staged via TENSOR_LOAD_TO_LDS (Tensor Data Mover), s_wait_tensorcnt.
// - Projections and aggregation on V_WMMA_F32_16X16X4_F32 (fp32, matches reference).
// - K-loops chunked (8 preloaded A/B pairs -> 8 back-to-back WMMAs) to amortize waits.

typedef __attribute__((ext_vector_type(2))) float v2f;
typedef __attribute__((ext_vector_type(8))) float v8f;
typedef __attribute__((ext_vector_type(4))) unsigned int v4u;
typedef __attribute__((ext_vector_type(8))) unsigned int v8u;

namespace {
constexpr int Bn = 64;    // batch
constexpr int Wd = 128;   // input feature dim (W)
constexpr int Fn = 64;    // nodes (F)
constexpr int Hn = 4;     // heads
constexpr int OW = 128;   // per-head output dim
constexpr float NEG = 0.2f;

constexpr int NFS = Wd + 4;         // 132: row stride for nf/fs/fd panels (pad vs banks)
constexpr int SCS = Fn + 2;         // 66:  row stride for score panels
constexpr int SEG   = Fn * NFS;     // 8448 floats per [64][132] panel
constexpr int SCSEG = Fn * SCS;     // 4224 floats per [64][66] panel

// fs panel is aliased with the raw x[b] staging buffer (8192 <= 8448).
constexpr int LDS_FLOATS = 3 * SEG + 2 * SCSEG + OW;   // 33,920 floats = 135,680 B
}

__global__ __launch_bounds__(256)
void gatv2_fused(const float* __restrict__ x,
                 const float* __restrict__ Wsrc, const float* __restrict__ bsrc,
                 const float* __restrict__ Wdst, const float* __restrict__ bdst,
                 const float* __restrict__ attn,
                 float* __restrict__ out)
{
  extern __shared__ float lds[];
  float* fsb = lds;                  // [Fn][NFS] fs panel; ALSO raw x[b] [w][f] during stage-in
  float* nfs = fsb + SEG;            // [Fn][NFS] nf[f][w] = x[b,w,f] (transposed, A-operand)
  float* fdb = nfs + SEG;            // [Fn][NFS] fd panel
  float* sc  = fdb + SEG;            // [Fn][SCS] raw scores[i][j]
  float* scT = sc  + SCSEG;          // [Fn][SCS] softmaxed, transposed: scT[j][i] = a[i][j]
  float* ah  = scT + SCSEG;          // [OW] attn[h][:]

  const int b = blockIdx.x >> 2;     // batch index
  const int h = blockIdx.x & 3;      // head index
  const int tid  = threadIdx.x;
  const int wave = tid >> 5;
  const int lane = tid & 31;
  const int ln    = lane & 15;       // N/M position within 16-wide tile
  const int kbase = (lane >> 4) << 1; // 0 or 2 : K pair owned by this lane half
  const int mhalf = (lane >> 4) << 3; // 0 or 8 : C/D row-half owned by this lane half

  float* xraw = fsb;                 // raw x[b] ([Wd][Fn], 8192 floats) staged here

  // ---- Stage x[b] (32 KB) into LDS via the Tensor Data Mover (wave 0 only) --
  if (wave == 0) {
    const unsigned long long ga =
        (unsigned long long)(uintptr_t)(x + (size_t)b * Wd * Fn);
    const unsigned int ldsOff = (unsigned int)(uintptr_t)xraw;  // LDS byte offset

    // D# group 0: count=1 | lds_addr | global_addr[56:0] | type=2
    v4u g0;
    g0[0] = 1u;                                        // count=1, user mode, no gather
    g0[1] = ldsOff;                                    // lds_addr (bytes)
    g0[2] = (unsigned int)(ga & 0xFFFFFFFFu);          // global_addr[31:0]
    g0[3] = ((unsigned int)(ga >> 32) & 0x01FFFFFFu)   // global_addr[56:32]
          | (2u << 30);                                // type = 2 ("image")
    // D# group 1: 2D tensor, data_size=4B, dim0=64 (f), dim1=128 (w), full tile.
    v8u g1;
    g1[0] = 2u << 16;                  // data_size = 2 (4 bytes); mask/flags = 0
    g1[1] = (64u & 0xFFFFu) << 16;     // tensor_dim0[15:0]=64 (atomic_barrier_addr=0)
    g1[2] = (128u << 16);              // tensor_dim0[31:16]=0 | tensor_dim1[15:0]=128
    g1[3] = (64u << 16);               // tensor_dim1[31:16]=0 | tile_dim0=64
    g1[4] = 128u;                      // tile_dim1=128 | tile_dim2=0
    g1[5] = 64u;                       // tensor_dim0_stride[31:0] = 64 elements
    g1[6] = 0u;                        // stride0[47:32]=0 | tensor_dim1_stride[15:0]=0
    g1[7] = 0u;
    asm volatile("tensor_load_to_lds %0, %1" :: "s"(g0), "s"(g1) : "memory");
    __builtin_amdgcn_s_wait_tensorcnt(0);
  }
  if (tid < OW) ah[tid] = attn[h * OW + tid];
  __syncthreads();

  // ---- Transpose raw x[b] ([w][f]) into nf panel ([f][w], stride NFS) ------
  for (int idx = tid; idx < Wd * Fn; idx += 256) {
    const int w = idx >> 6, f = idx & 63;
    nfs[f * NFS + w] = xraw[idx];
  }
  __syncthreads();   // xraw (== fs panel) is dead from here; phase 1 reuses it

  // ---- Phase 1: fs/fd projections via WMMA f32 16x16x4 ---------------------
  // fs[f,o] = sum_w nf[f][w] * Wsrc[h*OW+o][w] + bsrc[h*OW+o]   (same for fd)
  // 64 output tiles (32 fs + 32 fd), 8 waves -> 8 tiles/wave.  K=128 -> 32 steps,
  // chunked 8-deep: one load clause feeds 8 back-to-back WMMAs.
  for (int t8 = 0; t8 < 8; ++t8) {
    const int t = wave * 8 + t8;               // 0..63, wave-uniform
    const bool isSrc = (t < 32);
    const int tt = t & 31;
    const int mt = tt >> 3;                    // f-tile 0..3
    const int nt = tt & 7;                     // o-tile 0..7
    const float* Wm  = isSrc ? Wsrc : Wdst;
    const float* bia = isSrc ? bsrc : bdst;
    float* dst = isSrc ? fsb : fdb;

    const int ocol = nt * 16 + ln;             // this lane's output column
    const float* wrow = Wm + (size_t)(h * OW + ocol) * Wd;  // B[k][n] = W[h*OW+n][k]
    __builtin_prefetch(wrow, 0, 0);
    const float* arow = nfs + (mt * 16 + ln) * NFS;         // A[m][k] = nf[m][k]

    v8f c = {};
    #pragma unroll
    for (int ch = 0; ch < 4; ++ch) {
      v2f av[8], bv[8];
      #pragma unroll
      for (int u = 0; u < 8; ++u) {            // load clause: 8x ds_load_b64 + 8x global_load_b64
        const int k = (ch * 8 + u) * 4 + kbase;
        av[u] = *(const v2f*)(arow + k);
        bv[u] = *(const v2f*)(wrow + k);
      }
      #pragma unroll
      for (int u = 0; u < 8; ++u)              // dense WMMA issue
        c = __builtin_amdgcn_wmma_f32_16x16x4_f32(
              false, av[u], false, bv[u], (short)0, c, false, false);
    }
    const float bvv = bia[h * OW + ocol];
    #pragma unroll
    for (int r = 0; r < 8; ++r)                // D row M = mt*16 + mhalf + r, col N = ocol
      dst[(mt * 16 + mhalf + r) * NFS + ocol] = c[r] + bvv;
  }
  __syncthreads();

  // ---- Phase 2a: pairwise logits (irreducibly nonlinear -> VALU, float4) ----
  // scores[i,j] = sum_d attn[h,d] * lrelu(fs[i,d] + fd[j,d])
  {
    const int j  = tid & 63;
    const int i0 = tid >> 6;
    const float4* fj4 = (const float4*)(fdb + j * NFS);   // 528B rows: 16B aligned
    const float4* ah4 = (const float4*)ah;
    for (int p = 0; p < 16; ++p) {
      const int i = i0 + p * 4;
      const float4* fi4 = (const float4*)(fsb + i * NFS);
      float s = 0.0f;
      #pragma unroll 4
      for (int d4 = 0; d4 < OW / 4; ++d4) {
        const float4 aa = ah4[d4], u = fi4[d4], v = fj4[d4];
        float t0 = u.x + v.x; t0 = (t0 >= 0.0f) ? t0 : NEG * t0; s = fmaf(aa.x, t0, s);
        float t1 = u.y + v.y; t1 = (t1 >= 0.0f) ? t1 : NEG * t1; s = fmaf(aa.y, t1, s);
        float t2 = u.z + v.z; t2 = (t2 >= 0.0f) ? t2 : NEG * t2; s = fmaf(aa.z, t2, s);
        float t3 = u.w + v.w; t3 = (t3 >= 0.0f) ? t3 : NEG * t3; s = fmaf(aa.w, t3, s);
      }
      sc[i * SCS + j] = s;
    }
  }
  __syncthreads();

  // ---- Phase 2b: softmax over sources i per column j; write TRANSPOSED ------
  // scT[j][i] = a[i][j]  -> contiguous rows so phase 2c A-loads are ds_load_b64.
  if (tid < Fn) {
    const int j = tid;
    float mx = -3.4e38f;
    for (int i = 0; i < Fn; ++i) mx = fmaxf(mx, sc[i * SCS + j]);
    float sum = 0.0f;
    float* rowT = scT + j * SCS;
    for (int i = 0; i < Fn; ++i) {
      const float e = __expf(sc[i * SCS + j] - mx);
      rowT[i] = e;
      sum += e;
    }
    const float inv = 1.0f / sum;
    for (int i = 0; i < Fn; ++i) rowT[i] *= inv;
  }
  __syncthreads();

  // ---- Phase 2c: aggregation out_h = a^T @ fs via WMMA, K = 64 --------------
  // D[j,d] = sum_i a[i,j] * fs[i,d].  32 tiles, 8 waves -> 4 tiles/wave.
  float* ob = out + (size_t)b * OW * Fn;       // out layout [B][OW][Fn]: out[b, d, j]
  for (int t4 = 0; t4 < 4; ++t4) {
    const int t  = wave * 4 + t4;              // 0..31, wave-uniform
    const int jt = t >> 3;                     // j-tile 0..3
    const int dt = t & 7;                      // d-tile 0..7
    const int n  = dt * 16 + ln;               // B col (d)
    const float* arow = scT + (jt * 16 + ln) * SCS;  // A[m=j][k=i] = scT[j][i]

    v8f c = {};
    #pragma unroll
    for (int ch = 0; ch < 2; ++ch) {
      v2f av[8], bv[8];
      #pragma unroll
      for (int u = 0; u < 8; ++u) {
        const int k = (ch * 8 + u) * 4 + kbase;      // k = i
        av[u] = *(const v2f*)(arow + k);             // ds_load_b64
        bv[u].x = fsb[k * NFS + n];                  // B[i][d] = fs[i][d] (row pair)
        bv[u].y = fsb[(k + 1) * NFS + n];
      }
      #pragma unroll
      for (int u = 0; u < 8; ++u)
        c = __builtin_amdgcn_wmma_f32_16x16x4_f32(
              false, av[u], false, bv[u], (short)0, c, false, false);
    }
    #pragma unroll
    for (int r = 0; r < 8; ++r) {              // head-mean: += 0.25 * tile across 4 h-blocks
      const int j = jt * 16 + mhalf + r;
      atomicAdd(ob + (size_t)n * Fn + j, 0.25f * c[r]);
    }
  }
}

extern "C" void kernel_launch(void* const* d_in, const int* in_sizes, int n_in,
                              void* d_out, int out_size, void* d_ws, size_t ws_size,
                              hipStream_t stream) {
  const float* x    = (const float*)d_in[0];
  const float* Wsrc = (const float*)d_in[1];
  const float* bsrc = (const float*)d_in[2];
  const float* Wdst = (const float*)d_in[3];
  const float* bdst = (const float*)d_in[4];
  const float* attn = (const float*)d_in[5];
  float* out = (float*)d_out;

  // Head-mean is accumulated with atomics across the 4 head-blocks -> zero first.
  hipMemsetAsync(out, 0, (size_t)out_size * sizeof(float), stream);

  const size_t lds_bytes = (size_t)LDS_FLOATS * sizeof(float);  // ~132.5 KB of 320 KB
  gatv2_fused<<<dim3(Bn * Hn), dim3(256), lds_bytes, stream>>>(
      x, Wsrc, bsrc, Wdst, bdst, attn, out);
}